// NodeScorer_35124242547174
// MI455X (gfx1250) — compile-verified
//
#include <hip/hip_runtime.h>
#include <hip/hip_bf16.h>

typedef float v2f __attribute__((ext_vector_type(2)));
typedef float v8f __attribute__((ext_vector_type(8)));

#define DIN 128
#define DH  64
#define DOUT 32

// ---- hardware fp32 atomic add (global_atomic_add_f32, no CAS loop) ----
__device__ __forceinline__ void fadd(float* p, float v) {
    unsafeAtomicAdd(p, v);
}

// ---------------------------------------------------------------------
// degree / normalization
// ---------------------------------------------------------------------
__global__ void deg_init_k(float* deg, int n) {
    int i = blockIdx.x * blockDim.x + threadIdx.x;
    if (i < n) deg[i] = 1.0f;                 // self-loop
}

__global__ void deg_edges_k(const int* __restrict__ dst, float* deg, int e) {
    int i = blockIdx.x * blockDim.x + threadIdx.x;
    if (i < e) fadd(&deg[dst[i]], 1.0f);
}

__global__ void dinv_k(const float* __restrict__ deg, float* dinv, int n) {
    int i = blockIdx.x * blockDim.x + threadIdx.x;
    if (i < n) dinv[i] = rsqrtf(deg[i]);
}

// ---------------------------------------------------------------------
// WMMA fp32 GEMM:  C[nrows x NC] = A[nrows x K] * W[K x NC]   (no bias)
// One wave -> one 16x16 output tile, V_WMMA_F32_16X16X4_F32 over K.
// A layout per ISA: lanes 0-15 hold M=0..15 / K=k,k+1 ; lanes 16-31 K=k+2,k+3.
// B layout: lanes 0-15 hold N=0..15 / K=k,k+1 ; lanes 16-31 K=k+2,k+3.
// D layout: VGPR v -> row = v + 8*(lane>=16), col = lane&15.
// ---------------------------------------------------------------------
template<int K, int NC, int CT, int RT>
__global__ __launch_bounds__(32 * CT * RT)
void gemm_wmma_f32(const float* __restrict__ A, const float* __restrict__ W,
                   float* __restrict__ C, int nrows)
{
    const int lane = threadIdx.x & 31;
    const int wave = threadIdx.x >> 5;
    const int rt   = wave / CT;
    const int ct   = wave - rt * CT;
    const int row0 = blockIdx.x * (RT * 16) + rt * 16;

    const int half = lane >> 4;          // 0: K=k..k+1, 1: K=k+2..k+3
    const int m16  = lane & 15;
    const int koff = half * 2;

    int arow = row0 + m16;
    if (arow > nrows - 1) arow = nrows - 1;      // clamp tail (stores predicated)
    const float* Ap = A + (size_t)arow * K;

    const int n = ct * 16 + m16;                  // output column for this lane

    v8f acc = {};
#pragma unroll
    for (int k = 0; k < K; k += 4) {
        v2f a = *(const v2f*)(Ap + k + koff);     // 8B-aligned (k+koff even)
        v2f b;
        b.x = W[(k + koff)     * NC + n];
        b.y = W[(k + koff + 1) * NC + n];
        acc = __builtin_amdgcn_wmma_f32_16x16x4_f32(
            false, a, false, b, (short)0, acc, false, false);
    }

#pragma unroll
    for (int v = 0; v < 8; ++v) {
        int r = row0 + v + half * 8;
        if (r < nrows) C[(size_t)r * NC + n] = acc[v];
    }
}

// ---------------------------------------------------------------------
// agg init: self-loop term  agg = h * dinv^2
// ---------------------------------------------------------------------
template<int F>
__global__ void self_init_k(const float* __restrict__ h,
                            const float* __restrict__ dinv,
                            float* __restrict__ agg, int total)
{
    int idx = blockIdx.x * blockDim.x + threadIdx.x;
    if (idx >= total) return;
    int node = idx / F;
    float di = dinv[node];
    agg[idx] = h[idx] * di * di;
}

// ---------------------------------------------------------------------
// edge scatter: one wave per edge, lane = feature (coalesced 128B)
// ---------------------------------------------------------------------
__global__ void agg_edges64_k(const int* __restrict__ src, const int* __restrict__ dst,
                              const float* __restrict__ dinv,
                              const float* __restrict__ h, float* __restrict__ agg, int e)
{
    int gid  = blockIdx.x * blockDim.x + threadIdx.x;
    int edge = gid >> 5;
    int lane = gid & 31;
    if (edge >= e) return;
    int s = src[edge], d = dst[edge];
    float c = dinv[s] * dinv[d];
    const float* hs = h   + (size_t)s * 64;
    float*       ad = agg + (size_t)d * 64;
    fadd(ad + lane,      hs[lane]      * c);
    fadd(ad + lane + 32, hs[lane + 32] * c);
}

__global__ void agg_edges32_k(const int* __restrict__ src, const int* __restrict__ dst,
                              const float* __restrict__ dinv,
                              const float* __restrict__ h, float* __restrict__ agg, int e)
{
    int gid  = blockIdx.x * blockDim.x + threadIdx.x;
    int edge = gid >> 5;
    int lane = gid & 31;
    if (edge >= e) return;
    int s = src[edge], d = dst[edge];
    float c = dinv[s] * dinv[d];
    fadd(agg + (size_t)d * 32 + lane, h[(size_t)s * 32 + lane] * c);
}

// ---------------------------------------------------------------------
// bias + relu (in place)
// ---------------------------------------------------------------------
template<int F>
__global__ void bias_relu_k(float* __restrict__ agg, const float* __restrict__ b, int total)
{
    int idx = blockIdx.x * blockDim.x + threadIdx.x;
    if (idx >= total) return;
    agg[idx] = fmaxf(agg[idx] + b[idx & (F - 1)], 0.0f);
}

// ---------------------------------------------------------------------
// finalize: h = relu(agg2 + b2) -> d_out[N..], scores = h @ fc_w + fc_b
// one wave per node, 32 lanes = 32 features, wave32 xor-reduction
// ---------------------------------------------------------------------
__global__ void finalize_k(const float* __restrict__ agg2, const float* __restrict__ b2,
                           const float* __restrict__ fcw, const float* __restrict__ fcb,
                           float* __restrict__ scores, float* __restrict__ hout, int n)
{
    int gid  = blockIdx.x * blockDim.x + threadIdx.x;
    int node = gid >> 5;
    int lane = gid & 31;
    if (node >= n) return;
    float v = fmaxf(agg2[(size_t)node * 32 + lane] + b2[lane], 0.0f);
    hout[(size_t)node * 32 + lane] = v;
    float s = v * fcw[lane];
#pragma unroll
    for (int off = 16; off > 0; off >>= 1)
        s += __shfl_xor(s, off, 32);
    if (lane == 0) scores[node] = s + fcb[0];
}

// ---------------------------------------------------------------------
extern "C" void kernel_launch(void* const* d_in, const int* in_sizes, int n_in,
                              void* d_out, int out_size, void* d_ws, size_t ws_size,
                              hipStream_t stream)
{
    const float* x   = (const float*)d_in[0];
    const int*   ei  = (const int*)  d_in[1];
    const float* W1  = (const float*)d_in[2];
    const float* b1  = (const float*)d_in[3];
    const float* W2  = (const float*)d_in[4];
    const float* b2  = (const float*)d_in[5];
    const float* fcw = (const float*)d_in[6];
    const float* fcb = (const float*)d_in[7];

    const int N = in_sizes[0] / DIN;
    const int E = in_sizes[1] / 2;
    const int* src = ei;
    const int* dst = ei + E;

    // workspace carving (floats): deg[N] dinv[N] h1[N*64] agg1[N*64] h2[N*32]
    float* ws   = (float*)d_ws;
    float* deg  = ws;
    float* dinv = deg  + N;
    float* h1   = dinv + N;
    float* agg1 = h1   + (size_t)N * DH;   // also reused in-place as relu(h) layer-1 output
    float* h2   = agg1 + (size_t)N * DH;
    float* agg2 = h1;                      // h1 dead after GEMM2 -> reuse for agg2[N*32]

    float* scores = (float*)d_out;         // [N]
    float* hout   = scores + N;            // [N*32]

    const int T = 256;
    // 1) degrees + normalization
    deg_init_k <<<(N + T - 1) / T, T, 0, stream>>>(deg, N);
    deg_edges_k<<<(E + T - 1) / T, T, 0, stream>>>(dst, deg, E);
    dinv_k     <<<(N + T - 1) / T, T, 0, stream>>>(deg, dinv, N);

    // 2) layer 1: h1 = x @ W1  (WMMA, block = 8 waves = 2 row-tiles x 4 col-tiles)
    gemm_wmma_f32<DIN, DH, 4, 2><<<(N + 31) / 32, 256, 0, stream>>>(x, W1, h1, N);

    // 3) aggregate layer 1
    self_init_k<DH><<<((size_t)N * DH + T - 1) / T, T, 0, stream>>>(h1, dinv, agg1, N * DH);
    agg_edges64_k<<<((size_t)E * 32 + T - 1) / T, T, 0, stream>>>(src, dst, dinv, h1, agg1, E);
    bias_relu_k<DH><<<((size_t)N * DH + T - 1) / T, T, 0, stream>>>(agg1, b1, N * DH);

    // 4) layer 2: h2 = relu1 @ W2  (block = 8 waves = 4 row-tiles x 2 col-tiles)
    gemm_wmma_f32<DH, DOUT, 2, 4><<<(N + 63) / 64, 256, 0, stream>>>(agg1, W2, h2, N);

    // 5) aggregate layer 2
    self_init_k<DOUT><<<((size_t)N * DOUT + T - 1) / T, T, 0, stream>>>(h2, dinv, agg2, N * DOUT);
    agg_edges32_k<<<((size_t)E * 32 + T - 1) / T, T, 0, stream>>>(src, dst, dinv, h2, agg2, E);

    // 6) bias + relu + row-sum scores
    finalize_k<<<((size_t)N * 32 + T - 1) / T, T, 0, stream>>>(agg2, b2, fcw, fcb, scores, hout, N);
}